// Fusion_Nuclear_60335700574317
// MI455X (gfx1250) — compile-verified
//
#include <hip/hip_runtime.h>
#include <hip/hip_bf16.h>

// Polar Newton-Schulz nuclear-norm pooling for MI455X (gfx1250, wave32, WMMA).
//
//   X0 = (A + 1e-10) / ||A + 1e-10||_F           (all singular values <= 1)
//   X  <- 1.5*X - 0.5*X*(X^T X)   (NS_ITERS times; X -> polar factor U)
//   ||A||_* = <U, A + 1e-10>
//   out = (p1*ir + p2*vi) / (p1 + p2 + 1e-10)    (per-channel scalars)
//
// Round-2 changes:
//  * ns_gram exploits S = S^T: only 10 of 16 macro-tiles computed, mirrored.
//  * Both GEMM kernels double-buffer LDS via GLOBAL_LOAD_ASYNC_TO_LDS_B32/B128
//    (ASYNCcnt + s_wait_asynccnt), overlapping the global->LDS DMA of chunk
//    k+16 with the WMMA work on chunk k.
//
// Workspace layout (d_ws, requires 3*64MB + 1KB = ~192MB):
//   X  : 256 * 65536 f32   (ping)
//   X2 : 256 * 65536 f32   (pong)
//   S  : 256 * 65536 f32   (Gram temp)
//   p  : 256 f32           (nuclear norms; [0,128) = ir, [128,256) = vi)

typedef __attribute__((ext_vector_type(2))) float v2f;
typedef __attribute__((ext_vector_type(8))) float v8f;

#define NMAT   256
#define N      256
#define MELEMS 65536            // 256*256
#define NS_ITERS 30
#define LDSK   18               // K-stride, transposed panels (even -> 8B b64)
#define LDSKA  20               // K-stride, row-major A panel (80B = 5*16B)

__device__ __forceinline__ v2f lds_v2(const float* p) { return *(const v2f*)p; }

// ---- CDNA5 async global->LDS DMA (ASYNCcnt-tracked) -----------------------
__device__ __forceinline__ void async_ld_b32(uint32_t lds_off, const float* g) {
  asm volatile("global_load_async_to_lds_b32 %0, %1, off"
               :: "v"(lds_off), "v"((uint64_t)(uintptr_t)g) : "memory");
}
__device__ __forceinline__ void async_ld_b128(uint32_t lds_off, const float* g) {
  asm volatile("global_load_async_to_lds_b128 %0, %1, off"
               :: "v"(lds_off), "v"((uint64_t)(uintptr_t)g) : "memory");
}
__device__ __forceinline__ void wait_async0() {
  asm volatile("s_wait_asynccnt 0x0" ::: "memory");
}
__device__ __forceinline__ uint32_t lds_off32(const void* p) {
  return (uint32_t)(uintptr_t)p;   // LDS aperture: addr[31:0] = wave LDS offset
}

// ---------------------------------------------------------------- init ------
__global__ __launch_bounds__(256) void ns_init(const float* __restrict__ ir,
                                               const float* __restrict__ vi,
                                               float* __restrict__ X) {
  const int m = blockIdx.x;
  const float* A = (m < 128) ? (ir + (size_t)m * MELEMS)
                             : (vi + (size_t)(m - 128) * MELEMS);
  __shared__ float red[256];
  float s = 0.f;
  for (int i = threadIdx.x; i < MELEMS; i += 256) {
    float v = A[i] + 1e-10f;
    s += v * v;
  }
  red[threadIdx.x] = s;
  __syncthreads();
  for (int off = 128; off > 0; off >>= 1) {
    if (threadIdx.x < off) red[threadIdx.x] += red[threadIdx.x + off];
    __syncthreads();
  }
  const float inv = rsqrtf(red[0] + 1e-30f);
  float* Xm = X + (size_t)m * MELEMS;
  for (int i = threadIdx.x; i < MELEMS; i += 256)
    Xm[i] = (A[i] + 1e-10f) * inv;
}

// ------------------------------------------------------- S = X^T X ----------
// grid = (10, 256): upper-triangular 64x64 macro-tiles (S symmetric).
// 8 waves/block, 2 accumulator tiles per wave, V_WMMA_F32_16X16X4_F32.
__device__ __forceinline__ void gram_issue(const float* Xm,
                                           float (*As)[LDSK], float (*Bs)[LDSK],
                                           int kc, int i0, int j0, int tid) {
#pragma unroll
  for (int q = 0; q < 4; ++q) {
    const int e = tid + q * 256;        // 0..1023
    const int r = e >> 6, c = e & 63;   // k = r, col = c (coalesced in c)
    async_ld_b32(lds_off32(&As[c][r]), Xm + (size_t)(kc + r) * N + i0 + c);
    async_ld_b32(lds_off32(&Bs[c][r]), Xm + (size_t)(kc + r) * N + j0 + c);
  }
}

__global__ __launch_bounds__(256) void ns_gram(const float* __restrict__ Xin,
                                               float* __restrict__ S) {
  __shared__ float As[2][64][LDSK];  // As[b][i_local][k_local] = X[k][i0+i]
  __shared__ float Bs[2][64][LDSK];  // Bs[b][j_local][k_local] = X[k][j0+j]
  const int m  = blockIdx.y;
  const int ti = (int)((0x3221110000ULL >> (4 * blockIdx.x)) & 0xF);
  const int tj = (int)((0x3323213210ULL >> (4 * blockIdx.x)) & 0xF);
  const int i0 = ti * 64, j0 = tj * 64;
  const float* Xm = Xin + (size_t)m * MELEMS;

  const int tid  = threadIdx.x;
  const int lane = tid & 31;
  const int wave = tid >> 5;        // 0..7
  const int lm   = lane & 15;
  const int hl   = lane >> 4;       // half-wave select (K pairs 0,1 vs 2,3)
  const int t0   = wave * 2;        // wave owns tiles t0, t0+1 of 4x4 grid
  const int tr0 = (t0 >> 2) * 16,       tc0 = (t0 & 3) * 16;
  const int tr1 = ((t0 + 1) >> 2) * 16, tc1 = ((t0 + 1) & 3) * 16;

  v8f acc0 = {}, acc1 = {};

  gram_issue(Xm, As[0], Bs[0], 0, i0, j0, tid);
  int buf = 0;
  for (int kc = 0; kc < N; kc += 16, buf ^= 1) {
    wait_async0();      // chunk kc landed (this wave) ...
    __syncthreads();    // ... and every wave; everyone done reading buf^1
    if (kc + 16 < N) gram_issue(Xm, As[buf ^ 1], Bs[buf ^ 1], kc + 16, i0, j0, tid);
    float (*Ab)[LDSK] = As[buf];
    float (*Bb)[LDSK] = Bs[buf];
#pragma unroll
    for (int kk = 0; kk < 16; kk += 4) {
      const int kl = kk + hl * 2;       // per-lane K pair (ISA A/B layout)
      v2f a0 = lds_v2(&Ab[tr0 + lm][kl]);
      v2f b0 = lds_v2(&Bb[tc0 + lm][kl]);
      acc0 = __builtin_amdgcn_wmma_f32_16x16x4_f32(false, a0, false, b0,
                                                   (short)0, acc0, false, false);
      v2f a1 = lds_v2(&Ab[tr1 + lm][kl]);
      v2f b1 = lds_v2(&Bb[tc1 + lm][kl]);
      acc1 = __builtin_amdgcn_wmma_f32_16x16x4_f32(false, a1, false, b1,
                                                   (short)0, acc1, false, false);
    }
  }

  float* Sm = S + (size_t)m * MELEMS;
#pragma unroll
  for (int v = 0; v < 8; ++v) {         // C/D layout: row = v + 8*hl
    Sm[(size_t)(i0 + tr0 + v + hl * 8) * N + j0 + tc0 + lm] = acc0[v];
    Sm[(size_t)(i0 + tr1 + v + hl * 8) * N + j0 + tc1 + lm] = acc1[v];
  }
  if (i0 != j0) {                       // mirror into the lower triangle
#pragma unroll
    for (int v = 0; v < 8; ++v) {
      Sm[(size_t)(j0 + tc0 + lm) * N + i0 + tr0 + v + hl * 8] = acc0[v];
      Sm[(size_t)(j0 + tc1 + lm) * N + i0 + tr1 + v + hl * 8] = acc1[v];
    }
  }
}

// --------------------------------------- Xout = 1.5*Xin - 0.5*(Xin * S) -----
__device__ __forceinline__ void upd_issue(const float* Xm, const float* Sm,
                                          float (*As)[LDSKA], float (*Bs)[LDSK],
                                          int kc, int i0, int j0, int tid) {
  { // A panel: rows of Xin, contiguous in K -> one b128 DMA per thread
    const int i = tid >> 2, seg = (tid & 3) * 4;
    async_ld_b128(lds_off32(&As[i][seg]), Xm + (size_t)(i0 + i) * N + kc + seg);
  }
#pragma unroll
  for (int q = 0; q < 4; ++q) {         // B panel: rows of S (coalesced in j)
    const int e = tid + q * 256;
    const int r = e >> 6, c = e & 63;
    async_ld_b32(lds_off32(&Bs[c][r]), Sm + (size_t)(kc + r) * N + j0 + c);
  }
}

__global__ __launch_bounds__(256) void ns_update(const float* __restrict__ Xin,
                                                 const float* __restrict__ Smat,
                                                 float* __restrict__ Xout) {
  __shared__ float As[2][64][LDSKA]; // As[b][i_local][k_local] = Xin[i0+i][k]
  __shared__ float Bs[2][64][LDSK];  // Bs[b][j_local][k_local] = S[k][j0+j]
  const int m  = blockIdx.y;
  const int i0 = (blockIdx.x & 3) * 64;
  const int j0 = (blockIdx.x >> 2) * 64;
  const float* Xm = Xin  + (size_t)m * MELEMS;
  const float* Sm = Smat + (size_t)m * MELEMS;

  const int tid  = threadIdx.x;
  const int lane = tid & 31;
  const int wave = tid >> 5;
  const int lm   = lane & 15;
  const int hl   = lane >> 4;
  const int t0   = wave * 2;
  const int tr0 = (t0 >> 2) * 16,       tc0 = (t0 & 3) * 16;
  const int tr1 = ((t0 + 1) >> 2) * 16, tc1 = ((t0 + 1) & 3) * 16;

  v8f acc0 = {}, acc1 = {};

  upd_issue(Xm, Sm, As[0], Bs[0], 0, i0, j0, tid);
  int buf = 0;
  for (int kc = 0; kc < N; kc += 16, buf ^= 1) {
    wait_async0();
    __syncthreads();
    if (kc + 16 < N) upd_issue(Xm, Sm, As[buf ^ 1], Bs[buf ^ 1], kc + 16, i0, j0, tid);
    float (*Ab)[LDSKA] = As[buf];
    float (*Bb)[LDSK]  = Bs[buf];
#pragma unroll
    for (int kk = 0; kk < 16; kk += 4) {
      const int kl = kk + hl * 2;
      v2f a0 = lds_v2(&Ab[tr0 + lm][kl]);
      v2f b0 = lds_v2(&Bb[tc0 + lm][kl]);
      acc0 = __builtin_amdgcn_wmma_f32_16x16x4_f32(false, a0, false, b0,
                                                   (short)0, acc0, false, false);
      v2f a1 = lds_v2(&Ab[tr1 + lm][kl]);
      v2f b1 = lds_v2(&Bb[tc1 + lm][kl]);
      acc1 = __builtin_amdgcn_wmma_f32_16x16x4_f32(false, a1, false, b1,
                                                   (short)0, acc1, false, false);
    }
  }

  float* Xo = Xout + (size_t)m * MELEMS;
#pragma unroll
  for (int v = 0; v < 8; ++v) {
    {
      const size_t idx = (size_t)(i0 + tr0 + v + hl * 8) * N + j0 + tc0 + lm;
      Xo[idx] = 1.5f * Xm[idx] - 0.5f * acc0[v];
    }
    {
      const size_t idx = (size_t)(i0 + tr1 + v + hl * 8) * N + j0 + tc1 + lm;
      Xo[idx] = 1.5f * Xm[idx] - 0.5f * acc1[v];
    }
  }
}

// ------------------------------------------------- p[m] = <U, A + eps> ------
__global__ __launch_bounds__(256) void ns_dot(const float* __restrict__ ir,
                                              const float* __restrict__ vi,
                                              const float* __restrict__ U,
                                              float* __restrict__ p) {
  const int m = blockIdx.x;
  const float* A  = (m < 128) ? (ir + (size_t)m * MELEMS)
                              : (vi + (size_t)(m - 128) * MELEMS);
  const float* Um = U + (size_t)m * MELEMS;
  __shared__ float red[256];
  float s = 0.f;
  for (int i = threadIdx.x; i < MELEMS; i += 256)
    s += (A[i] + 1e-10f) * Um[i];
  red[threadIdx.x] = s;
  __syncthreads();
  for (int off = 128; off > 0; off >>= 1) {
    if (threadIdx.x < off) red[threadIdx.x] += red[threadIdx.x + off];
    __syncthreads();
  }
  if (threadIdx.x == 0) p[m] = red[0];
}

// -------------------------------------- out = w1*ir + w2*vi (float4) --------
__global__ __launch_bounds__(256) void ns_combine(const float* __restrict__ ir,
                                                  const float* __restrict__ vi,
                                                  const float* __restrict__ p,
                                                  float* __restrict__ out) {
  const size_t idx = (size_t)blockIdx.x * 256 + threadIdx.x;  // float4 index
  const int c = (int)(idx >> 14);                             // 16384 float4/chan
  const float p1 = p[c], p2 = p[c + 128];
  const float inv = 1.0f / (p1 + p2 + 1e-10f);
  const float w1 = p1 * inv, w2 = p2 * inv;
  const float4 a = ((const float4*)ir)[idx];
  const float4 b = ((const float4*)vi)[idx];
  float4 r;
  r.x = w1 * a.x + w2 * b.x;
  r.y = w1 * a.y + w2 * b.y;
  r.z = w1 * a.z + w2 * b.z;
  r.w = w1 * a.w + w2 * b.w;
  ((float4*)out)[idx] = r;
}

// ---------------------------------------------------------------------------
extern "C" void kernel_launch(void* const* d_in, const int* in_sizes, int n_in,
                              void* d_out, int out_size, void* d_ws, size_t ws_size,
                              hipStream_t stream) {
  const float* ir = (const float*)d_in[0];
  const float* vi = (const float*)d_in[1];
  float* out = (float*)d_out;

  float* X  = (float*)d_ws;                       // 64 MB
  float* X2 = X  + (size_t)NMAT * MELEMS;         // 64 MB
  float* S  = X2 + (size_t)NMAT * MELEMS;         // 64 MB
  float* p  = S  + (size_t)NMAT * MELEMS;         // 1 KB

  ns_init<<<NMAT, 256, 0, stream>>>(ir, vi, X);

  for (int t = 0; t < NS_ITERS; ++t) {
    const float* xin  = (t & 1) ? X2 : X;
    float*       xout = (t & 1) ? X : X2;
    ns_gram  <<<dim3(10, NMAT), 256, 0, stream>>>(xin, S);   // upper triangle
    ns_update<<<dim3(16, NMAT), 256, 0, stream>>>(xin, S, xout);
  }
  // NS_ITERS even -> final polar factor lands back in X

  ns_dot<<<NMAT, 256, 0, stream>>>(ir, vi, X, p);
  ns_combine<<<(128 * MELEMS / 4) / 256, 256, 0, stream>>>(ir, vi, p, out);
}